// Kernel_37950331028107
// MI455X (gfx1250) — compile-verified
//
#include <hip/hip_runtime.h>
#include <hip/hip_bf16.h>

typedef __attribute__((ext_vector_type(2))) float v2f;
typedef __attribute__((ext_vector_type(8))) float v8f;
typedef __attribute__((address_space(3))) char lds_char;

#define PAST   512
#define FUTURE 256
#define TOTAL  768
#define DDIM   128
#define BATCH  8
#define CDIM   128
#define PAD128 132   // 128-wide LDS tile row stride (mod 64 = 4 -> conflict-free)
#define PAD512 516   // 512-wide LDS tile row stride (mod 64 = 4)

// ---------------------------------------------------------------------------
// f32 WMMA step: D(16x16) += A(16x4) * B(4x16).
// lane t: m/n = t&15, half = t>>4; a[v]=A[m][2*half+v]; b[v]=B[2*half+v][n];
// c[r]=D[r+8*half][n].
// ---------------------------------------------------------------------------
__device__ inline v8f wmma_f32_k4(v2f a, v2f b, v8f c) {
    return __builtin_amdgcn_wmma_f32_16x16x4_f32(
        false, a, false, b, (short)0, c, false, false);
}

// Async bulk copy global -> LDS, 16B per lane, tracked by ASYNCcnt.
__device__ inline void async_b128(void* lds, const void* g) {
    asm volatile("global_load_async_to_lds_b128 %0, %1, off"
                 :: "v"((lds_char*)lds), "v"(g) : "memory");
}
__device__ inline void wait_async0() {
    asm volatile("s_wait_asynccnt 0" ::: "memory");
}

// Stage a contiguous 16 x 128 f32 tile into LDS with padded row stride PAD128.
// 256 threads: 16 threads/row, 8 floats (2x b128) each.
__device__ inline void stage_tile16x128(float* __restrict__ s,
                                        const float* __restrict__ g, int tid) {
    int row = tid >> 4, l16 = tid & 15;
    const float* gp = g + (size_t)row * 128 + l16 * 8;
    float*       lp = s + row * PAD128 + l16 * 8;
    async_b128(lp,     gp);
    async_b128(lp + 4, gp + 4);
}

// ---------------------------------------------------------------------------
// Kernel 1: per-(b,d) normalization over 768 timesteps (ddof=1), split into
// Xp (B,512,128) and Xf (B,256,128). grid=B, block=128 (one d per thread).
// ---------------------------------------------------------------------------
__global__ void k_normalize(const float* __restrict__ x,
                            float* __restrict__ xp, float* __restrict__ xf) {
    int b = blockIdx.x, d = threadIdx.x;
    const float* xb = x + (size_t)b * TOTAL * DDIM;
    float s = 0.f, ss = 0.f;
    for (int t = 0; t < TOTAL; ++t) {
        float v = xb[t * DDIM + d];
        s += v; ss += v * v;
    }
    float mean = s * (1.0f / TOTAL);
    float var  = (ss - (float)TOTAL * mean * mean) * (1.0f / (TOTAL - 1));
    float inv  = rsqrtf(var);
    for (int t = 0; t < TOTAL; ++t) {
        float v = (xb[t * DDIM + d] - mean) * inv;
        if (t < PAST) xp[((size_t)b * PAST + t) * DDIM + d] = v;
        else          xf[((size_t)b * FUTURE + (t - PAST)) * DDIM + d] = v;
    }
}

// ---------------------------------------------------------------------------
// Kernel 2: G = W * W^T (128x128). 64 blocks x 1 wave, one 16x16 tile each.
// ---------------------------------------------------------------------------
__global__ void k_wwt(const float* __restrict__ W, float* __restrict__ G) {
    int ti = blockIdx.x, i = ti >> 3, j = ti & 7;
    int t = threadIdx.x, lane = t & 15, half = t >> 4;
    const float* ar = W + (size_t)(16 * i + lane) * DDIM;
    const float* br = W + (size_t)(16 * j + lane) * DDIM;
    v8f acc = {};
    for (int k0 = 0; k0 < DDIM; k0 += 4) {
        v2f a, b;
        a.x = ar[k0 + 2 * half];  a.y = ar[k0 + 2 * half + 1];
        b.x = br[k0 + 2 * half];  b.y = br[k0 + 2 * half + 1];
        acc = wmma_f32_k4(a, b, acc);
    }
#pragma unroll
    for (int r = 0; r < 8; ++r)
        G[(size_t)(16 * i + r + 8 * half) * DDIM + 16 * j + lane] = acc[r];
}

// ---------------------------------------------------------------------------
// Kernel 3: C[M x 128] = A[M x 128] * G[128 x 128]. 8-wave blocks: the 16-row
// A strip is async-staged into LDS once, each wave computes one 16-col tile.
// grid = M/16, block = 256.
// ---------------------------------------------------------------------------
__global__ void k_mm_AG(const float* __restrict__ A, const float* __restrict__ G,
                        float* __restrict__ C) {
    __shared__ float sA[16 * PAD128];
    int tid = threadIdx.x, mi = blockIdx.x;
    stage_tile16x128(sA, A + (size_t)(16 * mi) * DDIM, tid);
    wait_async0();
    __syncthreads();
    int w = tid >> 5;                       // wave id = column tile
    int t = tid & 31, lane = t & 15, half = t >> 4;
    const float* arow = sA + lane * PAD128;
    v8f acc = {};
    for (int k0 = 0; k0 < DDIM; k0 += 4) {
        v2f a, b;
        a.x = arow[k0 + 2 * half];
        a.y = arow[k0 + 2 * half + 1];
        b.x = G[(size_t)(k0 + 2 * half + 0) * DDIM + 16 * w + lane];
        b.y = G[(size_t)(k0 + 2 * half + 1) * DDIM + 16 * w + lane];
        acc = wmma_f32_k4(a, b, acc);
    }
#pragma unroll
    for (int r = 0; r < 8; ++r)
        C[(size_t)(16 * mi + r + 8 * half) * DDIM + 16 * w + lane] = acc[r];
}

// ---------------------------------------------------------------------------
// Kernel 4: q[r] = dot(A[r,:], B[r,:]) over 128. One wave32 per row.
// ---------------------------------------------------------------------------
__global__ void k_rowdot(const float* __restrict__ A, const float* __restrict__ B,
                         float* __restrict__ q) {
    int r = blockIdx.x, lane = threadIdx.x;
    const float* a = A + (size_t)r * DDIM;
    const float* b = B + (size_t)r * DDIM;
    float s = 0.f;
    for (int i = lane; i < DDIM; i += 32) s += a[i] * b[i];
#pragma unroll
    for (int off = 16; off > 0; off >>= 1) s += __shfl_xor(s, off, 32);
    if (lane == 0) q[r] = s;
}

// ---------------------------------------------------------------------------
// Kernel 5: fused cross-GEMM + logits + mask.
// cross[f,p] = V[b,f,:] . Xp[b,p,:]; alpha = mask ? -(qp - 2*cross + qf)/sc^2
// 8-wave blocks: the 16-row V strip (shared by all p-tiles) is async-staged
// into LDS; wave w handles p-tile blockIdx.x*8+w.
// grid = (PAST/128, FUTURE/16, B), block = 256.
// ---------------------------------------------------------------------------
__global__ void k_alpha(const float* __restrict__ V, const float* __restrict__ Xp,
                        const float* __restrict__ qp, const float* __restrict__ qf,
                        const int* __restrict__ Amat, const float* __restrict__ smoothing,
                        float* __restrict__ alpha) {
    __shared__ float sV[16 * PAD128];
    int tid = threadIdx.x, fi = blockIdx.y, b = blockIdx.z;
    stage_tile16x128(sV, V + ((size_t)b * FUTURE + 16 * fi) * DDIM, tid);
    wait_async0();
    __syncthreads();
    int w = tid >> 5, pi = blockIdx.x * 8 + w;
    int t = tid & 31, lane = t & 15, half = t >> 4;
    const float* vrow = sV + lane * PAD128;
    const float* prow = Xp + ((size_t)b * PAST + 16 * pi + lane) * DDIM;
    __builtin_prefetch(prow, 0, 0);
    v8f acc = {};
    for (int k0 = 0; k0 < DDIM; k0 += 4) {
        v2f a, bb;
        a.x  = vrow[k0 + 2 * half];  a.y  = vrow[k0 + 2 * half + 1];
        bb.x = prow[k0 + 2 * half];  bb.y = prow[k0 + 2 * half + 1];
        acc = wmma_f32_k4(a, bb, acc);
    }
    float sm     = smoothing[0];
    float scale  = 0.01f / (1.0f + expf(-sm));   // sigmoid(sm)*0.01
    float invsc2 = 1.0f / (scale * scale);
    int   p      = 16 * pi + lane;
    float qpv    = qp[b * PAST + p];
#pragma unroll
    for (int r = 0; r < 8; ++r) {
        int   f   = 16 * fi + r + 8 * half;
        float val = -(qpv - 2.0f * acc[r] + qf[b * FUTURE + f]) * invsc2;
        int   msk = Amat[(size_t)p * TOTAL + PAST + f];
        alpha[((size_t)b * FUTURE + f) * PAST + p] = (msk == 0) ? -__builtin_inff() : val;
    }
}

// ---------------------------------------------------------------------------
// Kernel 6: softmax over p (512) per (b,f) row, in place. 256 threads/row.
// ---------------------------------------------------------------------------
__global__ void k_softmax(float* __restrict__ alpha) {
    __shared__ float red[256];
    float* a = alpha + (size_t)blockIdx.x * PAST;
    int t = threadIdx.x;
    float v0 = a[t], v1 = a[t + 256];
    red[t] = fmaxf(v0, v1);
    __syncthreads();
    for (int s = 128; s > 0; s >>= 1) {
        if (t < s) red[t] = fmaxf(red[t], red[t + s]);
        __syncthreads();
    }
    float mx = red[0];
    __syncthreads();
    float e0 = expf(v0 - mx), e1 = expf(v1 - mx);
    red[t] = e0 + e1;
    __syncthreads();
    for (int s = 128; s > 0; s >>= 1) {
        if (t < s) red[t] += red[t + s];
        __syncthreads();
    }
    float inv = 1.0f / red[0];
    a[t] = e0 * inv;
    a[t + 256] = e1 * inv;
}

// ---------------------------------------------------------------------------
// Kernel 7: yh[b] (256x128) = alpha[b] (256x512) * y[b] (512x128).
// 8-wave blocks: 16x512 alpha strip async-staged into LDS (pad 516), wave w
// computes c-tile w. grid = (FUTURE/16, B), block = 256.
// ---------------------------------------------------------------------------
__global__ void k_yh(const float* __restrict__ alpha, const float* __restrict__ y,
                     float* __restrict__ out) {
    __shared__ float sAl[16 * PAD512];
    int tid = threadIdx.x, fi = blockIdx.x, b = blockIdx.y;
    {   // stage 16 x 512 contiguous rows: 16 threads/row, 32 floats each
        int row = tid >> 4, l16 = tid & 15;
        const float* gp = alpha + ((size_t)b * FUTURE + 16 * fi + row) * PAST + l16 * 32;
        float*       lp = sAl + row * PAD512 + l16 * 32;
#pragma unroll
        for (int c = 0; c < 8; ++c)
            async_b128(lp + 4 * c, gp + 4 * c);
    }
    wait_async0();
    __syncthreads();
    int w = tid >> 5;                        // c-tile
    int t = tid & 31, lane = t & 15, half = t >> 4;
    const float* arow = sAl + lane * PAD512;
    const float* yb   = y + (size_t)b * PAST * CDIM;
    __builtin_prefetch(yb + 16 * w + lane, 0, 0);
    v8f acc = {};
    for (int k0 = 0; k0 < PAST; k0 += 4) {
        v2f a, bb;
        a.x  = arow[k0 + 2 * half];
        a.y  = arow[k0 + 2 * half + 1];
        bb.x = yb[(size_t)(k0 + 2 * half + 0) * CDIM + 16 * w + lane];
        bb.y = yb[(size_t)(k0 + 2 * half + 1) * CDIM + 16 * w + lane];
        acc = wmma_f32_k4(a, bb, acc);
    }
#pragma unroll
    for (int r = 0; r < 8; ++r)
        out[((size_t)b * FUTURE + 16 * fi + r + 8 * half) * CDIM + 16 * w + lane] = acc[r];
}

// ---------------------------------------------------------------------------
extern "C" void kernel_launch(void* const* d_in, const int* in_sizes, int n_in,
                              void* d_out, int out_size, void* d_ws, size_t ws_size,
                              hipStream_t stream) {
    const float* x  = (const float*)d_in[0];   // (8,768,128)
    const float* y  = (const float*)d_in[1];   // (8,512,128)
    const int*   Am = (const int*)  d_in[2];   // (768,768)
    const float* W  = (const float*)d_in[3];   // (128,128)
    const float* sm = (const float*)d_in[4];   // (1,)
    float* out = (float*)d_out;                // (8,256,128)

    float* ws = (float*)d_ws;
    float* Xp    = ws;                                   // 524288
    float* Xf    = Xp + (size_t)BATCH * PAST * DDIM;     // 262144
    float* G     = Xf + (size_t)BATCH * FUTURE * DDIM;   // 16384
    float* U     = G  + (size_t)DDIM * DDIM;             // 524288
    float* V     = U  + (size_t)BATCH * PAST * DDIM;     // 262144
    float* qp    = V  + (size_t)BATCH * FUTURE * DDIM;   // 4096
    float* qf    = qp + (size_t)BATCH * PAST;            // 2048
    float* alpha = qf + (size_t)BATCH * FUTURE;          // 1048576

    k_normalize<<<BATCH, DDIM, 0, stream>>>(x, Xp, Xf);
    k_wwt<<<64, 32, 0, stream>>>(W, G);
    k_mm_AG<<<BATCH * PAST / 16, 256, 0, stream>>>(Xp, G, U);
    k_mm_AG<<<BATCH * FUTURE / 16, 256, 0, stream>>>(Xf, G, V);
    k_rowdot<<<BATCH * PAST, 32, 0, stream>>>(Xp, U, qp);
    k_rowdot<<<BATCH * FUTURE, 32, 0, stream>>>(Xf, V, qf);
    k_alpha<<<dim3(PAST / 128, FUTURE / 16, BATCH), 256, 0, stream>>>(
        V, Xp, qp, qf, Am, sm, alpha);
    k_softmax<<<BATCH * FUTURE, 256, 0, stream>>>(alpha);
    k_yh<<<dim3(FUTURE / 16, BATCH), 256, 0, stream>>>(alpha, y, out);
}